// GeoCryoAIHybridGraphModel_25898652794889
// MI455X (gfx1250) — compile-verified
//
#include <hip/hip_runtime.h>
#include <hip/hip_bf16.h>

// B=128 nodes, T=128 steps, F=256, HID=128, HEADS=4
#define Bn   128
#define Tn   128
#define Fn   256
#define HIDn 128
#define NROWS (Bn * Tn)   // 16384

typedef __attribute__((ext_vector_type(2))) float v2f;
typedef __attribute__((ext_vector_type(8))) float v8f;

// ---------------------------------------------------------------------------
// Spatial graph preprocessing: CSR by destination + GCN normalization weights.
// Single block, deterministic (no atomics). E is small (~3k edges, 128 nodes).
// ---------------------------------------------------------------------------
__global__ __launch_bounds__(128) void build_csr_kernel(
    const int* __restrict__ sp, int E,
    int* __restrict__ row_ptr, int* __restrict__ colI,
    float* __restrict__ wE, float* __restrict__ selfw) {
  __shared__ float dinv_sh[Bn];
  __shared__ int   cnt_sh[Bn];
  __shared__ int   off_sh[Bn + 1];
  const int tid = threadIdx.x;
  const int* src = sp;
  const int* dst = sp + E;

  int cnt = 0;
  for (int e = 0; e < E; ++e)
    if (dst[e] == tid) ++cnt;
  cnt_sh[tid] = cnt;
  float di = rsqrtf((float)(cnt + 1));   // +1 self loop
  dinv_sh[tid] = di;
  selfw[tid] = di * di;
  __syncthreads();

  if (tid == 0) {
    int acc = 0;
    for (int d = 0; d < Bn; ++d) { off_sh[d] = acc; acc += cnt_sh[d]; }
    off_sh[Bn] = acc;
  }
  __syncthreads();

  int pos = off_sh[tid];
  for (int e = 0; e < E; ++e) {
    if (dst[e] == tid) {
      int s = src[e];
      colI[pos] = s;
      wE[pos] = dinv_sh[s] * dinv_sh[tid];
      ++pos;
    }
  }
  row_ptr[tid] = off_sh[tid];
  if (tid == 0) row_ptr[Bn] = off_sh[Bn];
}

// ---------------------------------------------------------------------------
// FP32 WMMA GEMM: C[M,N] = A[M,K] @ B[K,N], row-major.
// Block = 128 threads (4 waves) computing a 64x64 tile; each wave owns a
// 32x32 quadrant (2x2 WMMA 16x16x4 fragments).  A 64x16 K-panel of A
// (transposed to [k][m]) and a 16x64 panel of B are staged in LDS with
// coalesced b128 global loads.  LDS row stride 80 dwords => the two
// half-waves (K rows kk and kk+2) hit disjoint bank groups (2*80 % 64 = 32).
// Fragment layouts per CDNA5 ISA 7.12.2:
//   A 16x4 : lanes 0-15 rows M, VGPR0/1 = K {0,1} (half 0) or K {2,3} (half 1)
//   B 4x16 : lanes 0-15 cols N, VGPR0/1 = K {0,1} / {2,3} by half-wave
//   C 16x16: VGPR j holds row M=j (+8 for lanes 16-31), col N=lane&15
// ---------------------------------------------------------------------------
#define LDS_S 80   // dword stride between K-rows in LDS panels

__global__ __launch_bounds__(128) void gemm_wmma_f32(
    const float* __restrict__ A, const float* __restrict__ B,
    float* __restrict__ C, int M, int N, int K, int ldc) {
  __shared__ float lA[16 * LDS_S];   // [k][m], m in [0,64)
  __shared__ float lB[16 * LDS_S];   // [k][n], n in [0,64)

  const int tid  = threadIdx.x;
  const int lane = tid & 31;
  const int wv   = tid >> 5;       // wave 0..3
  const int wr   = wv >> 1;        // wave row (0..1): 32-row half of tile
  const int wc   = wv & 1;         // wave col (0..1): 32-col half of tile
  const int half = lane >> 4;      // half-wave
  const int l    = lane & 15;
  const int n0 = blockIdx.x * 64;
  const int m0 = blockIdx.y * 64;

  // staging maps (block-wide, coalesced b128)
  const int am = tid >> 2;         // A row within 32-row pass
  const int ak = (tid & 3) * 4;    // A k-quad
  const int bk = tid >> 4;         // B k-row within 8-row pass
  const int bn = (tid & 15) * 4;   // B col-quad

  v8f acc[2][2] = {};

  for (int k0 = 0; k0 < K; k0 += 16) {
    // A panel 64x16 -> lA[k][m] (transpose in registers)
    #pragma unroll
    for (int it = 0; it < 2; ++it) {
      int m = am + it * 32;
      const float4 va = *(const float4*)(A + (size_t)(m0 + m) * K + k0 + ak);
      lA[(ak + 0) * LDS_S + m] = va.x;
      lA[(ak + 1) * LDS_S + m] = va.y;
      lA[(ak + 2) * LDS_S + m] = va.z;
      lA[(ak + 3) * LDS_S + m] = va.w;
    }
    // B panel 16x64 -> lB[k][n]
    #pragma unroll
    for (int it = 0; it < 2; ++it) {
      int kk = bk + it * 8;
      *(float4*)(lB + kk * LDS_S + bn) =
          *(const float4*)(B + (size_t)(k0 + kk) * N + n0 + bn);
    }
    __syncthreads();

    #pragma unroll
    for (int kk = 0; kk < 16; kk += 4) {
      v2f a[2], b[2];
      #pragma unroll
      for (int r = 0; r < 2; ++r) {
        int row = wr * 32 + r * 16 + l;
        a[r].x = lA[(kk + 2 * half + 0) * LDS_S + row];
        a[r].y = lA[(kk + 2 * half + 1) * LDS_S + row];
      }
      #pragma unroll
      for (int c = 0; c < 2; ++c) {
        int col = wc * 32 + c * 16 + l;
        b[c].x = lB[(kk + 2 * half + 0) * LDS_S + col];
        b[c].y = lB[(kk + 2 * half + 1) * LDS_S + col];
      }
      #pragma unroll
      for (int r = 0; r < 2; ++r)
        #pragma unroll
        for (int c = 0; c < 2; ++c)
          acc[r][c] = __builtin_amdgcn_wmma_f32_16x16x4_f32(
              false, a[r], false, b[c], (short)0, acc[r][c], false, false);
    }
    __syncthreads();
  }

  #pragma unroll
  for (int r = 0; r < 2; ++r)
    #pragma unroll
    for (int c = 0; c < 2; ++c)
      #pragma unroll
      for (int j = 0; j < 8; ++j) {
        int row = m0 + wr * 32 + r * 16 + j + half * 8;
        int col = n0 + wc * 32 + c * 16 + l;
        C[(size_t)row * ldc + col] = acc[r][c][j];
      }
}

// ---------------------------------------------------------------------------
// GCN aggregation + bias + LayerNorm (+ optional ReLU), one block per (d,t)
// row, blockDim == channel count (128 or 256).
// ---------------------------------------------------------------------------
__global__ void gcn_agg_ln(
    const float* __restrict__ H, const int* __restrict__ row_ptr,
    const int* __restrict__ colI, const float* __restrict__ wE,
    const float* __restrict__ selfw,
    const float* __restrict__ bias, const float* __restrict__ g,
    const float* __restrict__ bb,
    float* __restrict__ out, int C, int ldo, int col_ofs, int do_relu) {
  __shared__ float s1[256], s2[256];
  const int row = blockIdx.x;          // d*T + t
  const int d = row >> 7, t = row & 127;
  const int c = threadIdx.x;

  float acc = selfw[d] * H[(size_t)row * C + c];
  const int e0 = row_ptr[d], e1 = row_ptr[d + 1];
  for (int e = e0; e < e1; ++e) {
    int s = colI[e];
    acc += wE[e] * H[((size_t)s * Tn + t) * C + c];
  }
  acc += bias[c];

  s1[c] = acc; s2[c] = acc * acc;
  __syncthreads();
  for (int st = C >> 1; st > 0; st >>= 1) {
    if (c < st) { s1[c] += s1[c + st]; s2[c] += s2[c + st]; }
    __syncthreads();
  }
  float mu  = s1[0] / (float)C;
  float var = fmaxf(s2[0] / (float)C - mu * mu, 0.f);
  float y = (acc - mu) * rsqrtf(var + 1e-5f) * g[c] + bb[c];
  if (do_relu) y = fmaxf(y, 0.f);
  out[(size_t)row * ldo + col_ofs + c] = y;
}

// ---------------------------------------------------------------------------
// GAT layer 1 (4 heads x 128): per-row attention scores es/ed.
// ---------------------------------------------------------------------------
__global__ __launch_bounds__(128) void gat1_scores(
    const float* __restrict__ H1, const float* __restrict__ asrc,
    const float* __restrict__ adst, float* __restrict__ es,
    float* __restrict__ ed) {
  __shared__ float s1[128], s2[128];
  const int row = blockIdx.x;
  const int tid = threadIdx.x;
  float v[4];
  #pragma unroll
  for (int h = 0; h < 4; ++h) v[h] = H1[(size_t)row * 512 + h * 128 + tid];
  for (int h = 0; h < 4; ++h) {
    s1[tid] = v[h] * asrc[h * 128 + tid];
    s2[tid] = v[h] * adst[h * 128 + tid];
    __syncthreads();
    for (int st = 64; st > 0; st >>= 1) {
      if (tid < st) { s1[tid] += s1[tid + st]; s2[tid] += s2[tid + st]; }
      __syncthreads();
    }
    if (tid == 0) { es[row * 4 + h] = s1[0]; ed[row * 4 + h] = s2[0]; }
    __syncthreads();
  }
}

// Temporal graph: node v's in-neighbors = window [v-5, v+5] clipped, incl self.
__global__ __launch_bounds__(128) void gat1_agg(
    const float* __restrict__ H1, const float* __restrict__ es,
    const float* __restrict__ ed, const float* __restrict__ bias,
    float* __restrict__ out) {
  __shared__ float alpha[4][11];
  const int row = blockIdx.x;
  const int b = row >> 7, v = row & 127;
  const int tid = threadIdx.x;
  const int u0 = max(0, v - 5), u1 = min(Tn - 1, v + 5);
  const int nn = u1 - u0 + 1;

  if (tid < 4) {
    const int h = tid;
    const float edv = ed[row * 4 + h];
    float e[11]; float m = -1e30f;
    for (int i = 0; i < nn; ++i) {
      float x = es[(b * Tn + u0 + i) * 4 + h] + edv;
      x = x > 0.f ? x : 0.2f * x;           // leaky_relu(0.2)
      e[i] = x; m = fmaxf(m, x);
    }
    float z = 0.f;
    for (int i = 0; i < nn; ++i) { float t = expf(e[i] - m); e[i] = t; z += t; }
    float inv = 1.f / z;
    for (int i = 0; i < nn; ++i) alpha[h][i] = e[i] * inv;
  }
  __syncthreads();

  #pragma unroll
  for (int k = 0; k < 4; ++k) {            // head k owns channels [k*128, ...)
    int c = k * 128 + tid;
    float acc = 0.f;
    for (int i = 0; i < nn; ++i)
      acc += alpha[k][i] * H1[((size_t)(b * Tn + u0 + i)) * 512 + c];
    acc += bias[c];
    out[(size_t)row * 512 + c] = acc > 0.f ? acc : expm1f(acc);   // ELU
  }
}

// GAT layer 2 (1 head x 256)
__global__ __launch_bounds__(256) void gat2_scores(
    const float* __restrict__ H2, const float* __restrict__ asrc,
    const float* __restrict__ adst, float* __restrict__ es,
    float* __restrict__ ed) {
  __shared__ float s1[256], s2[256];
  const int row = blockIdx.x;
  const int tid = threadIdx.x;
  float v = H2[(size_t)row * 256 + tid];
  s1[tid] = v * asrc[tid];
  s2[tid] = v * adst[tid];
  __syncthreads();
  for (int st = 128; st > 0; st >>= 1) {
    if (tid < st) { s1[tid] += s1[tid + st]; s2[tid] += s2[tid + st]; }
    __syncthreads();
  }
  if (tid == 0) { es[row] = s1[0]; ed[row] = s2[0]; }
}

__global__ __launch_bounds__(256) void gat2_agg(
    const float* __restrict__ H2, const float* __restrict__ es,
    const float* __restrict__ ed, const float* __restrict__ bias,
    float* __restrict__ out /* combined buffer, stride 512, cols 256..511 */) {
  __shared__ float alpha[11];
  const int row = blockIdx.x;
  const int b = row >> 7, v = row & 127;
  const int tid = threadIdx.x;
  const int u0 = max(0, v - 5), u1 = min(Tn - 1, v + 5);
  const int nn = u1 - u0 + 1;

  if (tid == 0) {
    const float edv = ed[row];
    float e[11]; float m = -1e30f;
    for (int i = 0; i < nn; ++i) {
      float x = es[b * Tn + u0 + i] + edv;
      x = x > 0.f ? x : 0.2f * x;
      e[i] = x; m = fmaxf(m, x);
    }
    float z = 0.f;
    for (int i = 0; i < nn; ++i) { float t = expf(e[i] - m); e[i] = t; z += t; }
    float inv = 1.f / z;
    for (int i = 0; i < nn; ++i) alpha[i] = e[i] * inv;
  }
  __syncthreads();

  float acc = 0.f;
  for (int i = 0; i < nn; ++i)
    acc += alpha[i] * H2[((size_t)(b * Tn + u0 + i)) * 256 + tid];
  acc += bias[tid];
  out[(size_t)row * 512 + 256 + tid] = acc;
}

// Fusion epilogue: + bias, LayerNorm, exact-erf GELU -> d_out
__global__ __launch_bounds__(256) void fuse_ln_gelu(
    const float* __restrict__ Hf, const float* __restrict__ bias,
    const float* __restrict__ g, const float* __restrict__ bb,
    float* __restrict__ out) {
  __shared__ float s1[256], s2[256];
  const int row = blockIdx.x;
  const int tid = threadIdx.x;
  float x = Hf[(size_t)row * 256 + tid] + bias[tid];
  s1[tid] = x; s2[tid] = x * x;
  __syncthreads();
  for (int st = 128; st > 0; st >>= 1) {
    if (tid < st) { s1[tid] += s1[tid + st]; s2[tid] += s2[tid + st]; }
    __syncthreads();
  }
  float mu  = s1[0] / 256.f;
  float var = fmaxf(s2[0] / 256.f - mu * mu, 0.f);
  float y = (x - mu) * rsqrtf(var + 1e-5f) * g[tid] + bb[tid];
  float ge = 0.5f * y * (1.f + erff(y * 0.70710678118654752f));
  out[(size_t)row * 256 + tid] = ge;
}

// ---------------------------------------------------------------------------
static inline size_t al256(size_t x) { return (x + 255) & ~(size_t)255; }

extern "C" void kernel_launch(void* const* d_in, const int* in_sizes, int n_in,
                              void* d_out, int out_size, void* d_ws, size_t ws_size,
                              hipStream_t stream) {
  (void)n_in; (void)out_size; (void)ws_size;
  const float* x0       = (const float*)d_in[0];   // [B,T,F]
  const int*   sp       = (const int*)d_in[2];     // [2,Es]
  const float* gcn_W1   = (const float*)d_in[4];
  const float* gcn_b1   = (const float*)d_in[5];
  const float* ln1_g    = (const float*)d_in[6];
  const float* ln1_b    = (const float*)d_in[7];
  const float* gcn_W2   = (const float*)d_in[8];
  const float* gcn_b2   = (const float*)d_in[9];
  const float* ln2_g    = (const float*)d_in[10];
  const float* ln2_b    = (const float*)d_in[11];
  const float* gcn_W3   = (const float*)d_in[12];
  const float* gcn_b3   = (const float*)d_in[13];
  const float* ln3_g    = (const float*)d_in[14];
  const float* ln3_b    = (const float*)d_in[15];
  const float* gat1_W   = (const float*)d_in[16];
  const float* gat1_as  = (const float*)d_in[17];
  const float* gat1_ad  = (const float*)d_in[18];
  const float* gat1_b   = (const float*)d_in[19];
  const float* gat2_W   = (const float*)d_in[20];
  const float* gat2_as  = (const float*)d_in[21];
  const float* gat2_ad  = (const float*)d_in[22];
  const float* gat2_b   = (const float*)d_in[23];
  const float* fus_W    = (const float*)d_in[24];
  const float* fus_b    = (const float*)d_in[25];
  const float* fus_g    = (const float*)d_in[26];
  const float* fus_bb   = (const float*)d_in[27];

  const int Es = in_sizes[2] / 2;
  char* w = (char*)d_ws;
  size_t o = 0;
  int*   row_ptr = (int*)(w + o);   o = al256(o + (Bn + 1) * 4);
  float* selfw   = (float*)(w + o); o = al256(o + Bn * 4);
  int*   colI    = (int*)(w + o);   o = al256(o + (size_t)Es * 4);
  float* wE      = (float*)(w + o); o = al256(o + (size_t)Es * 4);
  float* es1     = (float*)(w + o); o = al256(o + (size_t)NROWS * 4 * 4);
  float* ed1     = (float*)(w + o); o = al256(o + (size_t)NROWS * 4 * 4);
  float* es2     = (float*)(w + o); o = al256(o + (size_t)NROWS * 4);
  float* ed2     = (float*)(w + o); o = al256(o + (size_t)NROWS * 4);
  float* bufA    = (float*)(w + o); o = al256(o + (size_t)NROWS * 512 * 4);
  float* bufB    = (float*)(w + o); o = al256(o + (size_t)NROWS * 512 * 4);
  float* bufC    = (float*)(w + o); o = al256(o + (size_t)NROWS * 512 * 4);

  // Spatial graph CSR + normalization
  build_csr_kernel<<<1, 128, 0, stream>>>(sp, Es, row_ptr, colI, wE, selfw);

  // ---- Spatial GCN stack (batched over all T timesteps) ----
  // GCN1: [16384,256]@[256,128]
  gemm_wmma_f32<<<dim3(HIDn / 64, NROWS / 64), 128, 0, stream>>>(
      x0, gcn_W1, bufA, NROWS, HIDn, Fn, HIDn);
  gcn_agg_ln<<<NROWS, HIDn, 0, stream>>>(bufA, row_ptr, colI, wE, selfw,
      gcn_b1, ln1_g, ln1_b, bufB, HIDn, HIDn, 0, 1);
  // GCN2: [16384,128]@[128,128]
  gemm_wmma_f32<<<dim3(HIDn / 64, NROWS / 64), 128, 0, stream>>>(
      bufB, gcn_W2, bufA, NROWS, HIDn, HIDn, HIDn);
  gcn_agg_ln<<<NROWS, HIDn, 0, stream>>>(bufA, row_ptr, colI, wE, selfw,
      gcn_b2, ln2_g, ln2_b, bufB, HIDn, HIDn, 0, 1);
  // GCN3: [16384,128]@[128,256]  -> combined cols [0,256)
  gemm_wmma_f32<<<dim3(Fn / 64, NROWS / 64), 128, 0, stream>>>(
      bufB, gcn_W3, bufA, NROWS, Fn, HIDn, Fn);
  gcn_agg_ln<<<NROWS, Fn, 0, stream>>>(bufA, row_ptr, colI, wE, selfw,
      gcn_b3, ln3_g, ln3_b, bufC, Fn, 512, 0, 0);

  // ---- Temporal GAT stack (batched over all B samples) ----
  // GAT1: [16384,256]@[256,512]
  gemm_wmma_f32<<<dim3(512 / 64, NROWS / 64), 128, 0, stream>>>(
      x0, gat1_W, bufA, NROWS, 512, Fn, 512);
  gat1_scores<<<NROWS, 128, 0, stream>>>(bufA, gat1_as, gat1_ad, es1, ed1);
  gat1_agg<<<NROWS, 128, 0, stream>>>(bufA, es1, ed1, gat1_b, bufB);
  // GAT2: [16384,512]@[512,256] -> combined cols [256,512)
  gemm_wmma_f32<<<dim3(Fn / 64, NROWS / 64), 128, 0, stream>>>(
      bufB, gat2_W, bufA, NROWS, Fn, 512, Fn);
  gat2_scores<<<NROWS, 256, 0, stream>>>(bufA, gat2_as, gat2_ad, es2, ed2);
  gat2_agg<<<NROWS, 256, 0, stream>>>(bufA, es2, ed2, gat2_b, bufC);

  // ---- Fusion: [16384,512]@[512,256] + bias + LN + GELU ----
  gemm_wmma_f32<<<dim3(Fn / 64, NROWS / 64), 128, 0, stream>>>(
      bufC, fus_W, bufA, NROWS, Fn, 512, Fn);
  fuse_ln_gelu<<<NROWS, 256, 0, stream>>>(bufA, fus_b, fus_g, fus_bb,
      (float*)d_out);
}